// APACT_29076928594558
// MI455X (gfx1250) — compile-verified
//
#include <hip/hip_runtime.h>
#include <math.h>

#define NP     80
#define NPIX   6400
#define ND     32
#define NPARAM 256
#define NIMG   (NPARAM * ND)
#define LAMBDA 0.03f

// d_out layout (floats): [x: 204800][tf: 409600][params: 3][loss: 1]
#define OUT_X  0
#define OUT_TF 204800
#define OUT_P  614400
#define OUT_L  614403

typedef __attribute__((ext_vector_type(2))) float v2f;
typedef __attribute__((ext_vector_type(8))) float v8f;

__device__ __forceinline__ v8f wmma4(v2f a, v2f b, v8f c) {
  // D = A(16x4 f32) * B(4x16 f32) + C(16x16 f32)
  return __builtin_amdgcn_wmma_f32_16x16x4_f32(
      /*neg_a=*/false, a, /*neg_b=*/false, b,
      /*c_mod=*/(short)0, c, /*reuse_a=*/false, /*reuse_b=*/false);
}

// Complex 80x80 matmul: D = A * B. Row-major re/im planes.
// 8 waves per block; each wave owns 16x16 output tiles (25 tiles total).
__device__ void cmatmul80(const float* __restrict__ Are, const float* __restrict__ Aim,
                          const float* __restrict__ Bre, const float* __restrict__ Bim,
                          float* __restrict__ Dre, float* __restrict__ Dim) {
  const int lane = threadIdx.x & 31;
  const int wave = threadIdx.x >> 5;   // 0..7 (uniform per wave -> EXEC stays all-1)
  const int half = lane >> 4;
  const int l15  = lane & 15;
  for (int tile = wave; tile < 25; tile += 8) {
    const int mt = tile / 5, nt = tile % 5;
    v8f cre = {};
    v8f cim = {};
    const int arow = (mt * 16 + l15) * NP;
    const int bcol = nt * 16 + l15;
    for (int k0 = 0; k0 < NP; k0 += 4) {
      const int ka = k0 + 2 * half;
      v2f ar, ai, br, bi, nai;
      // A frag: lane<16 -> K=k0,k0+1 ; lane>=16 -> K=k0+2,k0+3 (ISA 16x4 f32 layout)
      ar.x = Are[arow + ka];          ar.y = Are[arow + ka + 1];
      ai.x = Aim[arow + ka];          ai.y = Aim[arow + ka + 1];
      // B frag mirrors A with N across lanes
      br.x = Bre[ka * NP + bcol];     br.y = Bre[(ka + 1) * NP + bcol];
      bi.x = Bim[ka * NP + bcol];     bi.y = Bim[(ka + 1) * NP + bcol];
      cre = wmma4(ar, br, cre);       // Re += Ar*Br
      cim = wmma4(ar, bi, cim);       // Im += Ar*Bi
      cim = wmma4(ai, br, cim);       // Im += Ai*Br
      nai.x = -ai.x; nai.y = -ai.y;   // f32 WMMA has no A-neg modifier
      cre = wmma4(nai, bi, cre);      // Re -= Ai*Bi
    }
    for (int r = 0; r < 8; ++r) {     // C layout: vgpr r -> rows r / r+8
      const int row = mt * 16 + r + 8 * half;
      const int col = nt * 16 + l15;
      Dre[row * NP + col] = cre[r];
      Dim[row * NP + col] = cim[r];
    }
  }
}

// X = Fl * X * Fl via temp T (both in LDS). Fl from global ws.
__device__ void cfft2(const float* Fre, const float* Fim,
                      float* Xre, float* Xim, float* Tre, float* Tim) {
  __syncthreads();
  cmatmul80(Fre, Fim, Xre, Xim, Tre, Tim);
  __syncthreads();
  cmatmul80(Tre, Tim, Fre, Fim, Xre, Xim);
  __syncthreads();
}

// ---------------- kernels ----------------

__global__ void k_init(float* FfwdRe, float* FfwdIm, float* FinvRe, float* FinvIm) {
  int i = blockIdx.x * blockDim.x + threadIdx.x;
  if (i >= NPIX) return;
  int j = i / NP, k = i % NP;
  float t = 6.28318530717958647692f * (float)((j * k) % NP) / (float)NP;
  float c = cosf(t), s = sinf(t);
  FfwdRe[i] = c;                 // exp(-2*pi*i*jk/N)
  FfwdIm[i] = -s;
  FinvRe[i] = c * (1.0f / NP);   // conj(F)/N per dimension
  FinvIm[i] = s * (1.0f / NP);
}

__global__ void k_ffty(const float* __restrict__ y,
                       const float* __restrict__ FfwdRe, const float* __restrict__ FfwdIm,
                       float* __restrict__ YfRe, float* __restrict__ YfIm) {
  __shared__ float Xre[NPIX], Xim[NPIX], Tre[NPIX], Tim[NPIX];
  const int d = blockIdx.x;
  for (int i = threadIdx.x; i < NPIX; i += blockDim.x) { Xre[i] = y[d * NPIX + i]; Xim[i] = 0.f; }
  cfft2(FfwdRe, FfwdIm, Xre, Xim, Tre, Tim);
  for (int i = threadIdx.x; i < NPIX; i += blockDim.x) {
    YfRe[d * NPIX + i] = Xre[i];
    YfIm[d * NPIX + i] = Xim[i];
  }
}

__global__ void k_loss(const float* __restrict__ y, const float* __restrict__ tfF,
                       const float* __restrict__ FfwdRe, const float* __restrict__ FfwdIm,
                       const float* __restrict__ FinvRe, const float* __restrict__ FinvIm,
                       const float* __restrict__ YfRe, const float* __restrict__ YfIm,
                       float* __restrict__ loss_part) {
  __shared__ float Xre[NPIX], Xim[NPIX], Tre[NPIX], Tim[NPIX];
  const int img = blockIdx.x;          // p*32 + d
  const int d = img & (ND - 1);
  const int t = threadIdx.x;

  const float* tf = tfF + (size_t)img * NPIX * 2;   // complex64 interleaved
  for (int i = t; i < NPIX; i += blockDim.x) { Xre[i] = tf[2 * i]; Xim[i] = tf[2 * i + 1]; }

  // psf0 = ifft2(TF)   (all ifftshifts cancel analytically; see derivation)
  cfft2(FinvRe, FinvIm, Xre, Xim, Tre, Tim);

  // psf = |psf0| / sum
  float part = 0.f;
  for (int i = t; i < NPIX; i += blockDim.x) {
    float m = sqrtf(Xre[i] * Xre[i] + Xim[i] * Xim[i]);
    Xre[i] = m; Xim[i] = 0.f; part += m;
  }
  __syncthreads();
  Tre[t] = part; __syncthreads();
  for (int off = 128; off > 0; off >>= 1) { if (t < off) Tre[t] += Tre[t + off]; __syncthreads(); }
  const float inv = 1.0f / Tre[0];
  __syncthreads();
  for (int i = t; i < NPIX; i += blockDim.x) Xre[i] *= inv;

  // H0 = fft2(psf)
  cfft2(FfwdRe, FfwdIm, Xre, Xim, Tre, Tim);

  // W = H0 .* Yf[d]
  const float* yr = YfRe + d * NPIX;
  const float* yi = YfIm + d * NPIX;
  for (int i = t; i < NPIX; i += blockDim.x) {
    float hr = Xre[i], hi = Xim[i];
    float wr = hr * yr[i] - hi * yi[i];
    float wi = hr * yi[i] + hi * yr[i];
    Xre[i] = wr; Xim[i] = wi;
  }

  // y_hat = |ifft2(W)| ; accumulate (y - y_hat)^2
  cfft2(FinvRe, FinvIm, Xre, Xim, Tre, Tim);
  part = 0.f;
  const float* yd = y + d * NPIX;
  for (int i = t; i < NPIX; i += blockDim.x) {
    float m = sqrtf(Xre[i] * Xre[i] + Xim[i] * Xim[i]);
    float df = yd[i] - m;
    part += df * df;
  }
  __syncthreads();
  Tre[t] = part; __syncthreads();
  for (int off = 128; off > 0; off >>= 1) { if (t < off) Tre[t] += Tre[t + off]; __syncthreads(); }
  if (t == 0) loss_part[img] = Tre[0];    // deterministic per-image partial
}

__global__ void k_argmin(const float* __restrict__ loss_part, const float* __restrict__ params,
                         float* __restrict__ out, int* __restrict__ idx_slot) {
  __shared__ float sv[256];
  __shared__ int   si[256];
  const int t = threadIdx.x;
  float s = 0.f;
  for (int d = 0; d < ND; ++d) s += loss_part[t * ND + d];   // fixed order -> deterministic
  s *= (1.0f / (float)(ND * NPIX));
  sv[t] = s; si[t] = t; __syncthreads();
  for (int off = 128; off > 0; off >>= 1) {
    if (t < off && sv[t + off] < sv[t]) { sv[t] = sv[t + off]; si[t] = si[t + off]; }
    __syncthreads();
  }
  if (t == 0) {
    const int idx = si[0];
    *idx_slot = idx;
    out[OUT_P + 0] = params[idx * 3 + 0];
    out[OUT_P + 1] = params[idx * 3 + 1];
    out[OUT_P + 2] = params[idx * 3 + 2];
    out[OUT_L]     = sv[0];
  }
}

__global__ void k_bestx(const float* __restrict__ tfF,
                        const float* __restrict__ FfwdRe, const float* __restrict__ FfwdIm,
                        const float* __restrict__ FinvRe, const float* __restrict__ FinvIm,
                        const float* __restrict__ YfRe, const float* __restrict__ YfIm,
                        const int* __restrict__ idx_slot, float* __restrict__ out) {
  __shared__ float Xre[NPIX], Xim[NPIX], Tre[NPIX], Tim[NPIX];
  const int d = blockIdx.x;
  const int t = threadIdx.x;
  const int idx = *idx_slot;
  const int img = idx * ND + d;

  const float* tf = tfF + (size_t)img * NPIX * 2;
  for (int i = t; i < NPIX; i += blockDim.x) { Xre[i] = tf[2 * i]; Xim[i] = tf[2 * i + 1]; }

  cfft2(FinvRe, FinvIm, Xre, Xim, Tre, Tim);          // psf0
  float part = 0.f;
  for (int i = t; i < NPIX; i += blockDim.x) {
    float m = sqrtf(Xre[i] * Xre[i] + Xim[i] * Xim[i]);
    Xre[i] = m; Xim[i] = 0.f; part += m;
  }
  __syncthreads();
  Tre[t] = part; __syncthreads();
  for (int off = 128; off > 0; off >>= 1) { if (t < off) Tre[t] += Tre[t + off]; __syncthreads(); }
  const float inv = 1.0f / Tre[0];
  __syncthreads();
  for (int i = t; i < NPIX; i += blockDim.x) Xre[i] *= inv;

  cfft2(FfwdRe, FfwdIm, Xre, Xim, Tre, Tim);          // H0

  // G = conj(H0) .* Yf / (|H0|^2 + lambda)
  const float* yr = YfRe + d * NPIX;
  const float* yi = YfIm + d * NPIX;
  for (int i = t; i < NPIX; i += blockDim.x) {
    float hr = Xre[i], hi = Xim[i];
    float den = hr * hr + hi * hi + LAMBDA;
    float gr = (hr * yr[i] + hi * yi[i]) / den;
    float gi = (hr * yi[i] - hi * yr[i]) / den;
    Xre[i] = gr; Xim[i] = gi;
  }

  cfft2(FinvRe, FinvIm, Xre, Xim, Tre, Tim);          // x = |ifft2(G)|
  for (int i = t; i < NPIX; i += blockDim.x)
    out[OUT_X + d * NPIX + i] = sqrtf(Xre[i] * Xre[i] + Xim[i] * Xim[i]);
}

__global__ void k_copytf(const float* __restrict__ tfF, const int* __restrict__ idx_slot,
                         float* __restrict__ out) {
  const int i = blockIdx.x * blockDim.x + threadIdx.x;
  const int n = ND * NPIX * 2;   // 409600 floats = 204800 complex
  if (i < n) out[OUT_TF + i] = tfF[(size_t)(*idx_slot) * n + i];
}

extern "C" void kernel_launch(void* const* d_in, const int* in_sizes, int n_in,
                              void* d_out, int out_size, void* d_ws, size_t ws_size,
                              hipStream_t stream) {
  const float* y      = (const float*)d_in[0];   // (1,32,80,80) f32
  const float* tfF    = (const float*)d_in[1];   // (256,32,80,80) c64 -> float pairs
  const float* params = (const float*)d_in[2];   // (256,3) f32
  float* out = (float*)d_out;
  float* ws  = (float*)d_ws;

  float* FfwdRe = ws;                 // 6400
  float* FfwdIm = ws + 6400;          // 6400
  float* FinvRe = ws + 12800;         // 6400
  float* FinvIm = ws + 19200;         // 6400
  float* YfRe   = ws + 25600;         // 204800
  float* YfIm   = ws + 230400;        // 204800
  float* lpart  = ws + 435200;        // 8192
  int*   idxs   = (int*)(ws + 443392);

  k_init  <<<(NPIX + 255) / 256, 256, 0, stream>>>(FfwdRe, FfwdIm, FinvRe, FinvIm);
  k_ffty  <<<ND,   256, 0, stream>>>(y, FfwdRe, FfwdIm, YfRe, YfIm);
  k_loss  <<<NIMG, 256, 0, stream>>>(y, tfF, FfwdRe, FfwdIm, FinvRe, FinvIm,
                                     YfRe, YfIm, lpart);
  k_argmin<<<1,    256, 0, stream>>>(lpart, params, out, idxs);
  k_bestx <<<ND,   256, 0, stream>>>(tfF, FfwdRe, FfwdIm, FinvRe, FinvIm,
                                     YfRe, YfIm, idxs, out);
  k_copytf<<<(ND * NPIX * 2 + 255) / 256, 256, 0, stream>>>(tfF, idxs, out);
}